// ClusterGNNWrapper_910533067387
// MI455X (gfx1250) — compile-verified
//
#include <hip/hip_runtime.h>

// ---------------------------------------------------------------------------
// ClusterGNN forward for MI455X (gfx1250), fp32 end-to-end.
// Pipeline (all on `stream`):
//   for l in 0..2:  zero(agg); scatter_edges(h_l -> agg); wmma_gemm -> h_{l+1}
//   zero(node_msg); fused edge-MLP (2 chained f32 WMMA GEMMs) + double atomic
//   scatter into node_msg; classifier (+residual) -> logits.
// Matrix math uses V_WMMA_F32_16X16X4_F32 (exact fp32; workload is L2/atomic
// bound so fp32 WMMA throughput is far more than enough).  The edge-feature
// gather uses GLOBAL_LOAD_ASYNC_TO_LDS_B128 (ASYNCcnt) so gathered data goes
// straight to LDS without a VGPR round trip.
// ---------------------------------------------------------------------------

#define N_NODES   25000
#define HID       128
#define N_EDGES   625000
#define EATTR     16
#define NCLS      4
#define INFEAT    (2 * HID + EATTR)            // 272
#define ROW_TILES ((N_NODES + 15) / 16)        // 1563
#define EDGE_TILES ((N_EDGES + 15) / 16)       // 39063
#define FSTRIDE   276   // 272 + 4 pad: rows stay 16B aligned (276*4 = 69*16),
                        // banks m*20 mod 64 distinct for m=0..15
#define ZSTRIDE   130   // 128 + 2 pad

typedef __attribute__((ext_vector_type(2))) float v2f;
typedef __attribute__((ext_vector_type(8))) float v8f;

static __device__ __forceinline__ v8f wmma4(v2f a, v2f b, v8f c) {
  // D = A(16x4 f32) * B(4x16 f32) + C(16x16 f32), wave32
  return __builtin_amdgcn_wmma_f32_16x16x4_f32(
      /*neg_a=*/false, a, /*neg_b=*/false, b,
      /*c_mod=*/(short)0, c, /*reuse_a=*/false, /*reuse_b=*/false);
}

static __device__ __forceinline__ void atomAddF(float* p, float v) {
  // relaxed agent-scope fadd -> global_atomic_add_f32 (no CAS loop)
  __hip_atomic_fetch_add(p, v, __ATOMIC_RELAXED, __HIP_MEMORY_SCOPE_AGENT);
}

// Async global -> LDS 16B copy, per-lane addresses; tracked by ASYNCcnt.
// ldsa is the 32-bit LDS byte offset (generic LDS address low 32 bits == LDS
// offset per the gfx1250 aperture rules).
static __device__ __forceinline__ void async_copy_b128(unsigned ldsa,
                                                       const float* gaddr) {
  asm volatile("global_load_async_to_lds_b128 %0, %1, off"
               :: "v"(ldsa), "v"(gaddr) : "memory");
}

static __device__ __forceinline__ void wait_async0() {
#if __has_builtin(__builtin_amdgcn_s_wait_asynccnt)
  __builtin_amdgcn_s_wait_asynccnt(0);
#else
  asm volatile("s_wait_asynccnt 0x0" ::: "memory");
#endif
}

// ---------------------------------------------------------------------------
__global__ __launch_bounds__(256) void zero_kernel(float4* __restrict__ p, int n4) {
  int i = blockIdx.x * 256 + threadIdx.x;
  if (i < n4) p[i] = make_float4(0.f, 0.f, 0.f, 0.f);
}

// ---------------------------------------------------------------------------
// agg[dst[e]] += h[src[e]] over all edges; 32 threads per edge, float4 chunks.
__global__ __launch_bounds__(256) void scatter_agg_kernel(
    const float* __restrict__ h, const int* __restrict__ src,
    const int* __restrict__ dst, float* __restrict__ agg) {
  int gid = blockIdx.x * 256 + threadIdx.x;       // E*32 = 20M threads
  int e = gid >> 5;
  if (e >= N_EDGES) return;
  int col = (gid & 31) * 4;
  int s = src[e], d = dst[e];
  float4 v = *(const float4*)(h + s * HID + col); // coalesced 512B per edge row
  float* ap = agg + d * HID + col;
  atomAddF(ap + 0, v.x);
  atomAddF(ap + 1, v.y);
  atomAddF(ap + 2, v.z);
  atomAddF(ap + 3, v.w);
}

// ---------------------------------------------------------------------------
// h_out = relu(h_in @ Wself + agg @ Wneigh + bias), [N,128] x [128,128] x2.
// Wave w of each block owns output column tile [16w,16w+16); its two weight
// column slices live in 128 VGPRs, loaded once; grid-stride over row tiles.
__global__ __launch_bounds__(256) void gemm_layer_kernel(
    const float* __restrict__ h_in, const float* __restrict__ agg,
    const float* __restrict__ Wself, const float* __restrict__ Wneigh,
    const float* __restrict__ bias, float* __restrict__ h_out) {
  const int lane  = threadIdx.x & 31;
  const int wave  = threadIdx.x >> 5;
  const int mlane = lane & 15;        // row within A-tile / col within B,C
  const int khalf = lane >> 4;        // selects K pair {0,1} vs {2,3}
  const int cbase = wave * 16;
  const int col   = cbase + mlane;

  v2f bs[32], bn[32];                 // B slices: [128 x 16] each, in regs
#pragma unroll
  for (int s = 0; s < 32; ++s) {
    int k = 4 * s + 2 * khalf;
    bs[s].x = Wself[k * HID + col];
    bs[s].y = Wself[(k + 1) * HID + col];
    bn[s].x = Wneigh[k * HID + col];
    bn[s].y = Wneigh[(k + 1) * HID + col];
  }
  const float bv = bias[col];

  for (int tile = blockIdx.x; tile < ROW_TILES; tile += gridDim.x) {
    int rbase = tile * 16;
    int rowA  = rbase + mlane;
    if (rowA >= N_NODES) rowA = N_NODES - 1;   // clamp; store is predicated
    const float* hp = h_in + rowA * HID;
    const float* ap = agg + rowA * HID;

    v8f c;
#pragma unroll
    for (int r = 0; r < 8; ++r) c[r] = bv;

#pragma unroll
    for (int s = 0; s < 32; ++s) {
      int k = 4 * s + 2 * khalf;
      v2f a1 = *(const v2f*)(hp + k);
      c = wmma4(a1, bs[s], c);
      v2f a2 = *(const v2f*)(ap + k);
      c = wmma4(a2, bn[s], c);
    }

#pragma unroll
    for (int r = 0; r < 8; ++r) {
      int m   = r + 8 * khalf;
      int row = rbase + m;
      if (row < N_NODES) {
        float v = c[r];
        h_out[row * HID + col] = v > 0.f ? v : 0.f;
      }
    }
  }
}

// ---------------------------------------------------------------------------
// Fused edge MLP + double scatter.  One block = 8 waves processes 16 edges per
// iteration: async-gather feat(16x272) straight into LDS; wave w computes z
// column tile [16w,16w+16) = relu(feat@W1+b1) (K=272, 68 WMMA), re-stripes
// through LDS, then msg tile = z@W2+b2 (K=128, 32 WMMA), and atomically
// scatters the tile to node_msg[src] and node_msg[dst].
__global__ __launch_bounds__(256) void edge_mlp_kernel(
    const float* __restrict__ h, const int* __restrict__ src,
    const int* __restrict__ dst, const float* __restrict__ eattr,
    const float* __restrict__ W1, const float* __restrict__ b1,
    const float* __restrict__ W2, const float* __restrict__ b2,
    float* __restrict__ node_msg) {
  __shared__ alignas(16) float featbuf[16 * FSTRIDE];   // 17.3 KB
  __shared__ alignas(16) float zbuf[16 * ZSTRIDE];      // 8.1 KB
  __shared__ int rowS[16], rowD[16];

  const int lane  = threadIdx.x & 31;
  const int wave  = threadIdx.x >> 5;
  const int mlane = lane & 15;
  const int khalf = lane >> 4;
  const int cbase = wave * 16;
  const int col   = cbase + mlane;

  v2f w1r[68];                              // W1 col slice [272 x 16], 136 VGPRs
#pragma unroll
  for (int s = 0; s < 68; ++s) {
    int k = 4 * s + 2 * khalf;
    w1r[s].x = W1[k * HID + col];
    w1r[s].y = W1[(k + 1) * HID + col];
  }
  v2f w2r[32];                              // W2 col slice [128 x 16], 64 VGPRs
#pragma unroll
  for (int s = 0; s < 32; ++s) {
    int k = 4 * s + 2 * khalf;
    w2r[s].x = W2[k * HID + col];
    w2r[s].y = W2[(k + 1) * HID + col];
  }
  const float b1v = b1[col];
  const float b2v = b2[col];

  for (int tile = blockIdx.x; tile < EDGE_TILES; tile += gridDim.x) {
    const int ebase = tile * 16;

    if (threadIdx.x < 16) {
      int e = ebase + threadIdx.x;
      if (e >= N_EDGES) e = N_EDGES - 1;
      rowS[threadIdx.x] = src[e];
      rowD[threadIdx.x] = dst[e];
    }
    __syncthreads();

    // cooperative async gather: feat = [h[src] | h[dst] | edge_attr],
    // 16 rows x 68 16B-chunks, global -> LDS direct (no VGPR round trip).
    // Padding rows (ebase+m >= E) gather clamped-but-valid data; their output
    // rows are simply never scattered, so no zero-fill is needed.
    for (int i = threadIdx.x; i < 16 * 68; i += 256) {
      int m = i / 68;          // edge row within tile
      int c = i - m * 68;      // 16B chunk within the 272-float row
      int e = ebase + m;
      if (e >= N_EDGES) e = N_EDGES - 1;
      const float* g;
      if (c < 32)      g = h + rowS[m] * HID + c * 4;
      else if (c < 64) g = h + rowD[m] * HID + (c - 32) * 4;
      else             g = eattr + (size_t)e * EATTR + (c - 64) * 4;
      unsigned ldsa = (unsigned)(uintptr_t)(&featbuf[m * FSTRIDE + c * 4]);
      async_copy_b128(ldsa, g);
    }
    wait_async0();             // this wave's async copies landed in LDS
    __syncthreads();           // ... and everyone else's too

    // stage 1: z = relu(feat @ W1 + b1)
    v8f acc;
#pragma unroll
    for (int r = 0; r < 8; ++r) acc[r] = b1v;
#pragma unroll
    for (int s = 0; s < 68; ++s) {
      int k = 4 * s + 2 * khalf;
      v2f a = *(const v2f*)(&featbuf[mlane * FSTRIDE + k]);
      acc = wmma4(a, w1r[s], acc);
    }
#pragma unroll
    for (int r = 0; r < 8; ++r) {           // relu + re-stripe C->A via LDS
      int m = r + 8 * khalf;
      float v = acc[r];
      zbuf[m * ZSTRIDE + col] = v > 0.f ? v : 0.f;
    }
    __syncthreads();

    // stage 2: msg = z @ W2 + b2
    v8f acc2;
#pragma unroll
    for (int r = 0; r < 8; ++r) acc2[r] = b2v;
#pragma unroll
    for (int s = 0; s < 32; ++s) {
      int k = 4 * s + 2 * khalf;
      v2f a = *(const v2f*)(&zbuf[mlane * ZSTRIDE + k]);
      acc2 = wmma4(a, w2r[s], acc2);
    }

    // double scatter-add
#pragma unroll
    for (int r = 0; r < 8; ++r) {
      int m = r + 8 * khalf;
      if (ebase + m < N_EDGES) {
        float v = acc2[r];
        atomAddF(&node_msg[rowS[m] * HID + col], v);
        atomAddF(&node_msg[rowD[m] * HID + col], v);
      }
    }
    __syncthreads();   // protect featbuf/zbuf/rowS before next iteration
  }
}

// ---------------------------------------------------------------------------
// logits = (h + node_msg) @ clf_w + clf_b ; [25000 x 4] output.
__global__ __launch_bounds__(256) void classifier_kernel(
    const float* __restrict__ h, const float* __restrict__ node_msg,
    const float* __restrict__ clf_w, const float* __restrict__ clf_b,
    float* __restrict__ out) {
  int id = blockIdx.x * 256 + threadIdx.x;
  if (id >= N_NODES * NCLS) return;
  int row = id >> 2;
  int c   = id & 3;
  float sum = clf_b[c];
  const float* hp = h + row * HID;
  const float* mp = node_msg + row * HID;
#pragma unroll 8
  for (int k = 0; k < HID; ++k) {
    sum += (hp[k] + mp[k]) * clf_w[k * NCLS + c];
  }
  out[id] = sum;
}

// ---------------------------------------------------------------------------
extern "C" void kernel_launch(void* const* d_in, const int* in_sizes, int n_in,
                              void* d_out, int out_size, void* d_ws, size_t ws_size,
                              hipStream_t stream) {
  const float* x      = (const float*)d_in[0];
  const int*   eidx   = (const int*)d_in[1];      // [2, E]
  const float* eattr  = (const float*)d_in[2];
  const float* wself  = (const float*)d_in[3];    // [3,128,128]
  const float* wneigh = (const float*)d_in[4];    // [3,128,128]
  const float* convb  = (const float*)d_in[5];    // [3,128]
  const float* w1     = (const float*)d_in[6];    // [272,128]
  const float* b1     = (const float*)d_in[7];
  const float* w2     = (const float*)d_in[8];    // [128,128]
  const float* b2     = (const float*)d_in[9];
  const float* clfw   = (const float*)d_in[10];   // [128,4]
  const float* clfb   = (const float*)d_in[11];
  float* out = (float*)d_out;

  const int* src = eidx;
  const int* dst = eidx + N_EDGES;

  // workspace: hA | hB | agg | node_msg  (4 x 12.8 MB)
  float* hA   = (float*)d_ws;
  float* hB   = hA + (size_t)N_NODES * HID;
  float* agg  = hB + (size_t)N_NODES * HID;
  float* nmsg = agg + (size_t)N_NODES * HID;

  const int n4        = N_NODES * HID / 4;               // 800000
  const int zeroGrid  = (n4 + 255) / 256;                // 3125
  const int scatGrid  = (N_EDGES * 32 + 255) / 256;      // 78125
  const int gemmGrid  = 512;
  const int mlpGrid   = 1024;
  const int clfGrid   = (N_NODES * NCLS + 255) / 256;    // 391

  const float* hcur = x;
  float* houts[3] = {hA, hB, hA};
  for (int l = 0; l < 3; ++l) {
    zero_kernel<<<zeroGrid, 256, 0, stream>>>((float4*)agg, n4);
    scatter_agg_kernel<<<scatGrid, 256, 0, stream>>>(hcur, src, dst, agg);
    gemm_layer_kernel<<<gemmGrid, 256, 0, stream>>>(
        hcur, agg, wself + (size_t)l * HID * HID, wneigh + (size_t)l * HID * HID,
        convb + (size_t)l * HID, houts[l]);
    hcur = houts[l];
  }
  // hcur == hA after layer 2
  zero_kernel<<<zeroGrid, 256, 0, stream>>>((float4*)nmsg, n4);
  edge_mlp_kernel<<<mlpGrid, 256, 0, stream>>>(hcur, src, dst, eattr,
                                               w1, b1, w2, b2, nmsg);
  classifier_kernel<<<clfGrid, 256, 0, stream>>>(hcur, nmsg, clfw, clfb, out);
}